// BPN_74191265071404
// MI455X (gfx1250) — compile-verified
//
#include <hip/hip_runtime.h>
#include <math.h>

// Belief-propagation diffusion for MI455X (gfx1250, wave32).
//  - priors GEMM uses V_WMMA_F32_16X16X4_F32 (f32 WMMA, 16x16 node tile per wave)
//  - edge kernel: one wave32 per reverse-edge pair; lanes 0-15 = edge i,
//    lanes 16-31 = rev[i]; rev-message exchange = shfl_xor(,16);
//    state dim = lane%16 -> coalesced 64B msg I/O, shfl reductions.
//  - scatter-add of log-messages via global f32 atomics into L2-resident accum.

typedef float v2f __attribute__((ext_vector_type(2)));
typedef float v8f __attribute__((ext_vector_type(8)));

#define STATES   16
#define FDIM     128
#define EPSF     1e-10f
#define EXP1F    2.7182818284590452f
#define EXP1M1F  1.7182818284590452f

__global__ void bpn_priors_wmma(const float* __restrict__ feat,
                                const float* __restrict__ W,
                                float* __restrict__ priors,
                                float* __restrict__ log_priors,
                                float* __restrict__ accum,
                                int N, int numTiles) {
  const int wave = blockIdx.x * (blockDim.x >> 5) + (threadIdx.x >> 5);
  if (wave >= numTiles) return;                 // wave-uniform; EXEC stays all-1
  const int lane = threadIdx.x & 31;
  const int half = lane >> 4;                   // 0: K={0,1} / M=r ; 1: K={2,3} / M=r+8
  const int s    = lane & 15;                   // state / column index

  int rowA = wave * 16 + s;                     // A-matrix row held by this lane
  if (rowA >= N) rowA = N - 1;                  // clamp tail (duplicates harmless)
  const float* frow = feat + (size_t)rowA * FDIM;

  v8f acc = {};
#pragma unroll
  for (int k0 = 0; k0 < FDIM; k0 += 4) {
    const int k = k0 + 2 * half;
    v2f a, b;
    a.x = frow[k];                              // A[16x4]: VGPR0=K, VGPR1=K+1
    a.y = frow[k + 1];
    b.x = W[(size_t)k * STATES + s];            // B[4x16]: row K striped over lanes
    b.y = W[(size_t)(k + 1) * STATES + s];
    acc = __builtin_amdgcn_wmma_f32_16x16x4_f32(false, a, false, b,
                                                (short)0, acc, false, false);
  }

#pragma unroll
  for (int r = 0; r < 8; ++r) {
    float logit = acc[r];
    // softmax over the 16 states (held across the 16 lanes of this half)
    float mx = logit;
    mx = fmaxf(mx, __shfl_xor(mx, 1, 32));
    mx = fmaxf(mx, __shfl_xor(mx, 2, 32));
    mx = fmaxf(mx, __shfl_xor(mx, 4, 32));
    mx = fmaxf(mx, __shfl_xor(mx, 8, 32));
    float ex = expf(logit - mx);
    float sm = ex;
    sm += __shfl_xor(sm, 1, 32);
    sm += __shfl_xor(sm, 2, 32);
    sm += __shfl_xor(sm, 4, 32);
    sm += __shfl_xor(sm, 8, 32);
    float pr = ex / sm;
    float lp = logf(fmaxf(pr, EPSF));
    const int rowO = wave * 16 + r + 8 * half;  // C/D layout: M = r + 8*half
    if (rowO < N) {
      const size_t o = (size_t)rowO * STATES + s;
      priors[o]     = pr;   // also serves as beliefs for iteration 1
      log_priors[o] = lp;
      accum[o]      = lp;   // seed scatter-add accumulator
    }
  }
}

__global__ void bpn_edge(const float* __restrict__ beliefs,
                         const int* __restrict__ src,
                         const int* __restrict__ dst,
                         const int* __restrict__ rev,
                         float* __restrict__ msgs,
                         float* __restrict__ accum,
                         int Eu, int firstIter) {
  const int p = blockIdx.x * (blockDim.x >> 5) + (threadIdx.x >> 5);
  if (p >= Eu) return;                          // wave-uniform guard
  const int lane = threadIdx.x & 31;
  const int half = lane >> 4;
  const int s    = lane & 15;

  const int j = rev[p];                         // partner (reverse) edge
  const int e = half ? j : p;                   // lanes 0-15: edge p, 16-31: edge j

  float mOld = firstIter ? 0.0625f : msgs[(size_t)e * STATES + s];
  const float mRev = __shfl_xor(mOld, 16, 32);  // reverse edge's old message

  const int   se = src[e];
  const float b  = beliefs[(size_t)se * STATES + s];
  const float x  = b / mRev;

  float t = x;                                  // row total over 16 states
  t += __shfl_xor(t, 1, 32);
  t += __shfl_xor(t, 2, 32);
  t += __shfl_xor(t, 4, 32);
  t += __shfl_xor(t, 8, 32);

  // potential = exp(eye): (x @ P)_s = total + (e-1)*x_s ; row sum = (15+e)*total
  const float m = (t + EXP1M1F * x) / ((15.0f + EXP1F) * t);
  msgs[(size_t)e * STATES + s] = m;             // in-place safe: wave owns the pair

  const float lm = logf(fmaxf(m, EPSF));
  const int   de = dst[e];
  atomicAdd(&accum[(size_t)de * STATES + s], lm);
}

__global__ void bpn_node(float* __restrict__ accum,
                         const float* __restrict__ log_priors,
                         float* __restrict__ beliefs_out,
                         int N) {
  const int w    = blockIdx.x * (blockDim.x >> 5) + (threadIdx.x >> 5);
  const int lane = threadIdx.x & 31;
  const int half = lane >> 4;
  const int s    = lane & 15;
  const int node = w * 2 + half;

  const bool ok = (node < N);
  const size_t o = (size_t)(ok ? node : 0) * STATES + s;
  float v = ok ? accum[o] : 0.0f;

  float mx = v;
  mx = fmaxf(mx, __shfl_xor(mx, 1, 32));
  mx = fmaxf(mx, __shfl_xor(mx, 2, 32));
  mx = fmaxf(mx, __shfl_xor(mx, 4, 32));
  mx = fmaxf(mx, __shfl_xor(mx, 8, 32));
  float ex = expf(v - mx);
  float sm = ex;
  sm += __shfl_xor(sm, 1, 32);
  sm += __shfl_xor(sm, 2, 32);
  sm += __shfl_xor(sm, 4, 32);
  sm += __shfl_xor(sm, 8, 32);

  if (ok) {
    beliefs_out[o] = ex / sm;
    accum[o]       = log_priors[o];             // reset for next iteration
  }
}

extern "C" void kernel_launch(void* const* d_in, const int* in_sizes, int n_in,
                              void* d_out, int out_size, void* d_ws, size_t ws_size,
                              hipStream_t stream) {
  const float* feat = (const float*)d_in[0];
  const float* W    = (const float*)d_in[1];
  const int*   src  = (const int*)d_in[2];
  const int*   dst  = (const int*)d_in[3];
  const int*   rev  = (const int*)d_in[4];

  const int N  = in_sizes[0] / FDIM;            // 100000
  const int E2 = in_sizes[2];                   // 3,200,000 directed edges
  const int Eu = E2 / 2;                        // reverse-edge pairs

  float* priors      = (float*)d_out;           // [N,16]
  float* beliefs_out = priors + (size_t)N * STATES;

  // workspace: 3 node buffers + 1 message buffer (~224 MB total)
  float* lp    = (float*)d_ws;                  // log_priors  [N*16]
  float* accum = lp    + (size_t)N * STATES;    // scatter-add [N*16]
  float* bel   = accum + (size_t)N * STATES;    // beliefs     [N*16]
  float* msgs  = bel   + (size_t)N * STATES;    // messages    [E2*16]

  const dim3 blk(256);                          // 8 wave32 per block
  const int tiles   = (N + 15) / 16;
  const int gBlocks = (tiles + 7) / 8;
  const int eBlocks = (Eu + 7) / 8;
  const int nWaves  = (N + 1) / 2;
  const int nBlocks = (nWaves + 7) / 8;

  bpn_priors_wmma<<<gBlocks, blk, 0, stream>>>(feat, W, priors, lp, accum, N, tiles);

  // iteration 1 (uniform initial messages, beliefs = priors)
  bpn_edge<<<eBlocks, blk, 0, stream>>>(priors, src, dst, rev, msgs, accum, Eu, 1);
  bpn_node<<<nBlocks, blk, 0, stream>>>(accum, lp, bel, N);
  // iteration 2
  bpn_edge<<<eBlocks, blk, 0, stream>>>(bel, src, dst, rev, msgs, accum, Eu, 0);
  bpn_node<<<nBlocks, blk, 0, stream>>>(accum, lp, bel, N);
  // iteration 3 (write beliefs straight to d_out)
  bpn_edge<<<eBlocks, blk, 0, stream>>>(bel, src, dst, rev, msgs, accum, Eu, 0);
  bpn_node<<<nBlocks, blk, 0, stream>>>(accum, lp, beliefs_out, N);
}